// PaiNNInteraction_22333829940092
// MI455X (gfx1250) — compile-verified
//
#include <hip/hip_runtime.h>
#include <hip/hip_bf16.h>
#include <math.h>

// ---------------------------------------------------------------------------
// PaiNN interaction for MI455X (gfx1250, wave32).
// Phase 1: RMSNorm + GLU-MLP with V_WMMA_F32_16X16X4_F32 (fp32-exact matmul).
// Phase 2: bandwidth-bound streaming reduction over the 100 MB Wij tensor.
// ---------------------------------------------------------------------------

typedef __attribute__((ext_vector_type(2))) float v2f;
typedef __attribute__((ext_vector_type(8))) float v8f;

#define DD      128
#define TWO_D   256
#define THREE_D 384
#define NN      128
#define BB      4
#define ROWS    (BB * NN)        // 512
#define EPSF    1e-6f

// ============================================================================
// Kernel 1: per-row RMSNorm -> GEMM1(128->256) -> GLU -> GEMM2(128->384) + bias
// One block = one 16-row M tile. 128 threads = 4 waves.
//   wave w computes N-tiles {4w..4w+3} of GEMM1 and {6w..6w+5} of GEMM2.
// ============================================================================
__global__ __launch_bounds__(128) void painn_ffn_kernel(
    const float* __restrict__ q, const float* __restrict__ norm_w,
    const float* __restrict__ W1, const float* __restrict__ b1,
    const float* __restrict__ W2, const float* __restrict__ b2,
    float* __restrict__ xbuf)
{
    __shared__ float h_lds[16][DD + 4];      // rmsnorm output, later h2 (GLU out)
    __shared__ float y_lds[16][TWO_D + 4];   // GEMM1 output

    const int row0 = blockIdx.x * 16;        // global row of tile
    const int tid  = threadIdx.x;
    const int wave = tid >> 5;               // 0..3
    const int lane = tid & 31;               // 0..31 (wave32)

    // ---- RMSNorm: wave w handles rows 4w..4w+3; each lane covers 4 elements.
    for (int rr = 0; rr < 4; ++rr) {
        const int r = wave * 4 + rr;
        const float* qr = q + (size_t)(row0 + r) * DD;
        float v0 = qr[lane], v1 = qr[lane + 32], v2 = qr[lane + 64], v3 = qr[lane + 96];
        float ss = v0 * v0 + v1 * v1 + v2 * v2 + v3 * v3;
        #pragma unroll
        for (int off = 16; off > 0; off >>= 1) ss += __shfl_xor(ss, off, 32);
        const float scale = rsqrtf(ss * (1.0f / 128.0f) + EPSF);
        h_lds[r][lane]      = v0 * scale * (1.0f + norm_w[lane]);
        h_lds[r][lane + 32] = v1 * scale * (1.0f + norm_w[lane + 32]);
        h_lds[r][lane + 64] = v2 * scale * (1.0f + norm_w[lane + 64]);
        h_lds[r][lane + 96] = v3 * scale * (1.0f + norm_w[lane + 96]);
    }
    __syncthreads();

    // A-fragment addressing (16x4 f32): lanes 0-15 -> K {0,1}; lanes 16-31 -> K {2,3}
    const int m     = lane & 15;
    const int khalf = (lane >> 4) * 2;       // 0 or 2
    const int crow  = (lane >> 4) << 3;      // C rows: +0 (lanes 0-15) / +8 (16-31)

    // ---- GEMM1: y = h @ W1^T  (N = 256 -> 16 tiles, 4 per wave)
    #pragma unroll
    for (int t = 0; t < 4; ++t) {
        const int ntile = wave * 4 + t;
        const int n     = ntile * 16 + (lane & 15);
        const float* w1n = W1 + (size_t)n * DD;
        v8f acc = {};
        #pragma unroll 4
        for (int k0 = 0; k0 < DD; k0 += 4) {
            v2f a, b;
            a[0] = h_lds[m][k0 + khalf];
            a[1] = h_lds[m][k0 + khalf + 1];
            b[0] = w1n[k0 + khalf];
            b[1] = w1n[k0 + khalf + 1];
            acc = __builtin_amdgcn_wmma_f32_16x16x4_f32(
                false, a, false, b, (short)0, acc, false, false);
        }
        #pragma unroll
        for (int r = 0; r < 8; ++r)
            y_lds[crow + r][ntile * 16 + (lane & 15)] = acc[r];
    }
    __syncthreads();

    // ---- GLU: h2 = silu(y[:, :128] + b1[:128]) * sigmoid(y[:, 128:] + b1[128:])
    for (int idx = tid; idx < 16 * DD; idx += 128) {
        const int r = idx >> 7, d = idx & 127;
        const float a = y_lds[r][d] + b1[d];
        const float g = y_lds[r][d + DD] + b1[d + DD];
        const float sa = a / (1.0f + __expf(-a));       // silu(a)
        const float sg = 1.0f / (1.0f + __expf(-g));    // sigmoid(g)
        h_lds[r][d] = sa * sg;
    }
    __syncthreads();

    // ---- GEMM2: x = h2 @ W2^T + b2  (N = 384 -> 24 tiles, 6 per wave)
    #pragma unroll
    for (int t = 0; t < 6; ++t) {
        const int ntile = wave * 6 + t;
        const int n     = ntile * 16 + (lane & 15);
        const float* w2n = W2 + (size_t)n * DD;
        v8f acc = {};
        #pragma unroll 4
        for (int k0 = 0; k0 < DD; k0 += 4) {
            v2f a, b;
            a[0] = h_lds[m][k0 + khalf];
            a[1] = h_lds[m][k0 + khalf + 1];
            b[0] = w2n[k0 + khalf];
            b[1] = w2n[k0 + khalf + 1];
            acc = __builtin_amdgcn_wmma_f32_16x16x4_f32(
                false, a, false, b, (short)0, acc, false, false);
        }
        const float bias = b2[n];
        #pragma unroll
        for (int r = 0; r < 8; ++r)
            xbuf[(size_t)(row0 + crow + r) * THREE_D + ntile * 16 + (lane & 15)]
                = acc[r] + bias;
    }
}

// ============================================================================
// Kernel 2: streaming pairwise reduction. One block per (b,i); thread = one d.
//   dq[d]     = sum_j Wij[.,j,d]      * x[j,d]      * mask[j]
//   dmu[c][d] = sum_j Wij[.,j,D+d]*x[j,D+d]*mask * dir[j,c]
//             + sum_j Wij[.,j,2D+d]*x[j,2D+d]*mask * mu[b,j,c,d]
// Wij streams from HBM once (~100 MB => ~4.4 us floor); x/mu stay in L2.
// ============================================================================
__global__ __launch_bounds__(128) void painn_interact_kernel(
    const float* __restrict__ q, const float* __restrict__ mu,
    const float* __restrict__ Wij, const float* __restrict__ dir_ij,
    const float* __restrict__ mask_ij, const float* __restrict__ xbuf,
    float* __restrict__ out_q, float* __restrict__ out_mu)
{
    const int bi = blockIdx.x;           // b*N + i
    const int b  = bi >> 7;
    const int d  = threadIdx.x;          // 0..127

    const float* __restrict__ wrow = Wij     + (size_t)bi * NN * THREE_D;
    const float* __restrict__ mrow = mask_ij + (size_t)bi * NN;
    const float* __restrict__ drow = dir_ij  + (size_t)bi * NN * 3;
    const float* __restrict__ xb   = xbuf    + (size_t)b * NN * THREE_D;
    const float* __restrict__ mub  = mu      + (size_t)b * NN * 3 * DD;

    float dq = 0.0f, a0 = 0.0f, a1 = 0.0f, a2 = 0.0f;

    for (int j = 0; j < NN; ++j) {
        const float* __restrict__ w  = wrow + (size_t)j * THREE_D;
        // keep the HBM stream ahead of the FMAs (gfx1250 global_prefetch_b8)
        __builtin_prefetch((const void*)(w + 8 * THREE_D), 0, 0);

        const float  mk  = mrow[j];
        const float  d0  = drow[j * 3 + 0];
        const float  d1  = drow[j * 3 + 1];
        const float  d2  = drow[j * 3 + 2];
        const float* __restrict__ x  = xb  + (size_t)j * THREE_D;
        const float* __restrict__ mj = mub + (size_t)j * 3 * DD;

        const float t0 = w[d]          * x[d]          * mk;
        const float t1 = w[DD + d]     * x[DD + d]     * mk;
        const float t2 = w[2 * DD + d] * x[2 * DD + d] * mk;

        dq += t0;
        a0 = fmaf(t1, d0, fmaf(t2, mj[d],          a0));
        a1 = fmaf(t1, d1, fmaf(t2, mj[DD + d],     a1));
        a2 = fmaf(t1, d2, fmaf(t2, mj[2 * DD + d], a2));
    }

    out_q[(size_t)bi * DD + d] = q[(size_t)bi * DD + d] + dq;

    const float* __restrict__ mui = mu + (size_t)bi * 3 * DD;
    float* __restrict__ om = out_mu + (size_t)bi * 3 * DD;
    om[d]          = mui[d]          + a0;
    om[DD + d]     = mui[DD + d]     + a1;
    om[2 * DD + d] = mui[2 * DD + d] + a2;
}

// ============================================================================
extern "C" void kernel_launch(void* const* d_in, const int* in_sizes, int n_in,
                              void* d_out, int out_size, void* d_ws, size_t ws_size,
                              hipStream_t stream) {
    const float* q       = (const float*)d_in[0];
    const float* mu      = (const float*)d_in[1];
    const float* Wij     = (const float*)d_in[2];
    const float* dir_ij  = (const float*)d_in[3];
    const float* mask_ij = (const float*)d_in[4];
    const float* norm_w  = (const float*)d_in[5];
    const float* W1      = (const float*)d_in[6];
    const float* b1      = (const float*)d_in[7];
    const float* W2      = (const float*)d_in[8];
    const float* b2      = (const float*)d_in[9];

    float* out_q  = (float*)d_out;                 // (B,N,D)
    float* out_mu = out_q + (size_t)BB * NN * DD;  // (B,N,3,D)
    float* xbuf   = (float*)d_ws;                  // (B*N, 3D) = 786 KB scratch

    painn_ffn_kernel<<<ROWS / 16, 128, 0, stream>>>(q, norm_w, W1, b1, W2, b2, xbuf);
    painn_interact_kernel<<<ROWS, 128, 0, stream>>>(q, mu, Wij, dir_ij, mask_ij,
                                                    xbuf, out_q, out_mu);
}